// KaolinRenderer_9320079032640
// MI455X (gfx1250) — compile-verified
//
#include <hip/hip_runtime.h>
#include <hip/hip_bf16.h>

#define H_IMG 240
#define W_IMG 320
#define BIG_NEG -1000000000.0f
#define NF_PAD 512

typedef float v2f __attribute__((ext_vector_type(2)));
typedef float v8f __attribute__((ext_vector_type(8)));

static __device__ inline v2f mk2(float a, float b) { v2f r; r.x = a; r.y = b; return r; }

// ---------------------------------------------------------------------------
// Workspace layout (pre-swizzled WMMA B-frags, built by geom_kernel):
//   v2f plane[4][1024]:  plane q, face f:
//       [q*1024 +        f] = (Aq, Bq)   -> read by lanes 0..15  (rows K0,K1)
//       [q*1024 + 512  + f] = (Cq, 0 )   -> read by lanes 16..31 (rows K2,K3)
//   planes: q=0 -> w0, q=1 -> w1, q=2 -> w2, q=3 -> z
//   float lam[512] at float-offset 8192.
// Total: 8704 floats = 34816 bytes.
// ---------------------------------------------------------------------------

__global__ __launch_bounds__(512) void geom_kernel(
    const float* __restrict__ verts, const int* __restrict__ faces,
    const float* __restrict__ Robj,  const float* __restrict__ Tobj,
    const float* __restrict__ Kc,    const float* __restrict__ Rw2c,
    const float* __restrict__ tw2c,  float* __restrict__ ws,
    int nv, int nf)
{
    __shared__ float sndc[NF_PAD * 3];   // per-vertex NDC (x,y,z)
    __shared__ float svcam[NF_PAD * 3];  // per-vertex camera-space position
    const int tid = threadIdx.x;

    // ---- build VP = P(K) @ world2cam (tiny, redundant per thread) ----
    const float fx = Kc[0], fy = Kc[4], cxk = Kc[2], cyk = Kc[5];
    float P[4][4] = {
        { 2.0f * fx / W_IMG, 0.0f, (W_IMG - 2.0f * cxk) / W_IMG, 0.0f },
        { 0.0f, 2.0f * fy / H_IMG, (H_IMG - 2.0f * cyk) / H_IMG, 0.0f },
        { 0.0f, 0.0f, -1.0f, -0.2f },
        { 0.0f, 0.0f, -1.0f,  0.0f }
    };
    float M[4][4];
    #pragma unroll
    for (int r = 0; r < 3; ++r) {
        M[r][0] = Rw2c[r * 3 + 0]; M[r][1] = Rw2c[r * 3 + 1];
        M[r][2] = Rw2c[r * 3 + 2]; M[r][3] = tw2c[r];
    }
    M[3][0] = 0.0f; M[3][1] = 0.0f; M[3][2] = 0.0f; M[3][3] = 1.0f;
    float VP[4][4];
    #pragma unroll
    for (int i = 0; i < 4; ++i)
        #pragma unroll
        for (int j = 0; j < 4; ++j)
            VP[i][j] = P[i][0] * M[0][j] + P[i][1] * M[1][j] +
                       P[i][2] * M[2][j] + P[i][3] * M[3][j];

    // ---- per-vertex: world transform, camera transform, projection ----
    if (tid < nv && tid < NF_PAD) {
        float v0 = verts[tid * 3 + 0], v1 = verts[tid * 3 + 1], v2 = verts[tid * 3 + 2];
        float wx = Robj[0] * v0 + Robj[1] * v1 + Robj[2] * v2 + Tobj[0];
        float wy = Robj[3] * v0 + Robj[4] * v1 + Robj[5] * v2 + Tobj[1];
        float wz = Robj[6] * v0 + Robj[7] * v1 + Robj[8] * v2 + Tobj[2];
        float vc[4];
        #pragma unroll
        for (int i = 0; i < 4; ++i)
            vc[i] = VP[i][0] * wx + VP[i][1] * wy + VP[i][2] * wz + VP[i][3];
        float vz = fmaxf(vc[2], 1e-8f);
        float inv = 1.0f / vz;
        sndc[tid * 3 + 0] = vc[0] * inv;
        sndc[tid * 3 + 1] = vc[1] * inv;
        sndc[tid * 3 + 2] = vc[2] * inv;
        svcam[tid * 3 + 0] = Rw2c[0] * wx + Rw2c[1] * wy + Rw2c[2] * wz + tw2c[0];
        svcam[tid * 3 + 1] = Rw2c[3] * wx + Rw2c[4] * wy + Rw2c[5] * wz + tw2c[1];
        svcam[tid * 3 + 2] = Rw2c[6] * wx + Rw2c[7] * wy + Rw2c[8] * wz + tw2c[2];
    }
    __syncthreads();

    // ---- per-face: three barycentric planes, depth plane, lambert ----
    float A0 = 0.f, B0 = 0.f, C0 = 0.f;
    float A1 = 0.f, B1 = 0.f, C1 = 0.f;
    float Az = 0.f, Bz = 0.f, Cz = BIG_NEG, lam = 0.f;
    if (tid < nf) {
        int i0 = faces[tid * 3 + 0], i1 = faces[tid * 3 + 1], i2 = faces[tid * 3 + 2];
        i0 = (i0 < 0) ? 0 : (i0 >= nv ? nv - 1 : i0);
        i1 = (i1 < 0) ? 0 : (i1 >= nv ? nv - 1 : i1);
        i2 = (i2 < 0) ? 0 : (i2 >= nv ? nv - 1 : i2);
        float ax = sndc[i0 * 3 + 0], ay = sndc[i0 * 3 + 1], z0 = sndc[i0 * 3 + 2];
        float bx = sndc[i1 * 3 + 0], by = sndc[i1 * 3 + 1], z1 = sndc[i1 * 3 + 2];
        float cx = sndc[i2 * 3 + 0], cy = sndc[i2 * 3 + 1], z2 = sndc[i2 * 3 + 2];
        float area = (bx - ax) * (cy - ay) - (by - ay) * (cx - ax);
        bool valid = fabsf(area) > 1e-8f;
        float inv = valid ? 1.0f / area : 0.0f;
        // w0 = ((cx-bx)(Py-by) - (cy-by)(Px-bx)) / area -> A0*Px + B0*Py + C0
        A0 = -(cy - by) * inv;
        B0 = (cx - bx) * inv;
        C0 = ((cy - by) * bx - (cx - bx) * by) * inv;
        // w1 = ((ax-cx)(Py-cy) - (ay-cy)(Px-cx)) / area
        A1 = -(ay - cy) * inv;
        B1 = (ax - cx) * inv;
        C1 = ((ay - cy) * cx - (ax - cx) * cy) * inv;
        // zp = z2 + w0*(z0-z2) + w1*(z1-z2)
        float dz0 = z0 - z2, dz1 = z1 - z2;
        Az = A0 * dz0 + A1 * dz1;
        Bz = B0 * dz0 + B1 * dz1;
        Cz = valid ? (z2 + C0 * dz0 + C1 * dz1) : BIG_NEG;
        // lambert from camera-space face normal
        float e1x = svcam[i1 * 3 + 0] - svcam[i0 * 3 + 0];
        float e1y = svcam[i1 * 3 + 1] - svcam[i0 * 3 + 1];
        float e1z = svcam[i1 * 3 + 2] - svcam[i0 * 3 + 2];
        float e2x = svcam[i2 * 3 + 0] - svcam[i0 * 3 + 0];
        float e2y = svcam[i2 * 3 + 1] - svcam[i0 * 3 + 1];
        float e2z = svcam[i2 * 3 + 2] - svcam[i0 * 3 + 2];
        float nx = e1y * e2z - e1z * e2y;
        float ny = e1z * e2x - e1x * e2z;
        float nz = e1x * e2y - e1y * e2x;
        float nrm = sqrtf(nx * nx + ny * ny + nz * nz) + 1e-8f;
        lam = fmaxf(nz / nrm, 0.0f);
        if (!valid) { A0 = B0 = C0 = A1 = B1 = C1 = Az = Bz = 0.f; Cz = BIG_NEG; }
    }
    // w2 plane = 1 - w0 - w1 (affine too)
    const float A2 = -(A0 + A1), B2 = -(B0 + B1), C2 = 1.0f - C0 - C1;

    v2f* wp = (v2f*)ws;
    wp[0 * 1024 +       tid] = mk2(A0, B0);
    wp[0 * 1024 + 512 + tid] = mk2(C0, 0.0f);
    wp[1 * 1024 +       tid] = mk2(A1, B1);
    wp[1 * 1024 + 512 + tid] = mk2(C1, 0.0f);
    wp[2 * 1024 +       tid] = mk2(A2, B2);
    wp[2 * 1024 + 512 + tid] = mk2(C2, 0.0f);
    wp[3 * 1024 +       tid] = mk2(Az, Bz);
    wp[3 * 1024 + 512 + tid] = mk2(Cz, 0.0f);
    ws[8192 + tid] = lam;
}

// ---------------------------------------------------------------------------
// Kernel 2: WMMA rasterizer. Each wave owns 16 consecutive pixels (16 | 320,
// so strips never straddle a row). Per 16-face tile: four
// V_WMMA_F32_16X16X4_F32 evaluate w0, w1, w2, zp for the 16x16 tile; the
// inside test is a single min3 >= 0, the z-race two cndmasks.
// ---------------------------------------------------------------------------
__global__ __launch_bounds__(256) void raster_kernel(
    const float* __restrict__ ws, float* __restrict__ out)
{
    __shared__ v2f   spair[4 * 1024];   // 32 KB pre-swizzled B-frags
    __shared__ float slam[NF_PAD];      //  2 KB lambert table
    const int tid = threadIdx.x;

    __builtin_prefetch(ws, 0, 1);                 // global_prefetch_b8
    const v2f* g2 = (const v2f*)ws;
    #pragma unroll
    for (int i = tid; i < 4 * 1024; i += 256) spair[i] = g2[i];
    #pragma unroll
    for (int i = tid; i < NF_PAD; i += 256) slam[i] = ws[8192 + i];
    __syncthreads();

    const int lane = tid & 31;
    const int wave = tid >> 5;
    const int pixBase = (blockIdx.x * 8 + wave) * 16;
    const int m = lane & 15;                      // pixel row (A) / face col (B)
    const bool lo = lane < 16;

    // A-frag (16x4 f32): lanes 0-15 hold K0,K1 = (Px,Py); lanes 16-31 hold
    // K2,K3 = (1,0) for pixel M = lane-16.
    const int p = pixBase + m;
    const int px = p % W_IMG, py = p / W_IMG;
    const float Px = ((float)px + 0.5f) * (2.0f / W_IMG) - 1.0f;
    const float Py = 1.0f - ((float)py + 0.5f) * (2.0f / H_IMG);
    v2f afrag;
    afrag.x = lo ? Px : 1.0f;
    afrag.y = lo ? Py : 0.0f;

    float bestZ[8], bestL[8];
    #pragma unroll
    for (int r = 0; r < 8; ++r) { bestZ[r] = BIG_NEG; bestL[r] = 0.0f; }

    const int idxBase = (lo ? 0 : 512) + m;       // lane slot in a plane
    const v8f zero = {};

    for (int t = 0; t < NF_PAD / 16; ++t) {
        const int idx = idxBase + t * 16;
        const v2f b0 = spair[0 * 1024 + idx];     // one ds_load_b64 per frag
        const v2f b1 = spair[1 * 1024 + idx];
        const v2f b2 = spair[2 * 1024 + idx];
        const v2f bz = spair[3 * 1024 + idx];
        const float lam = slam[t * 16 + m];       // this lane's face lambert

        v8f w0 = __builtin_amdgcn_wmma_f32_16x16x4_f32(
            false, afrag, false, b0, (short)0, zero, false, false);
        v8f w1 = __builtin_amdgcn_wmma_f32_16x16x4_f32(
            false, afrag, false, b1, (short)0, zero, false, false);
        v8f w2 = __builtin_amdgcn_wmma_f32_16x16x4_f32(
            false, afrag, false, b2, (short)0, zero, false, false);
        v8f zp = __builtin_amdgcn_wmma_f32_16x16x4_f32(
            false, afrag, false, bz, (short)0, zero, false, false);

        #pragma unroll
        for (int r = 0; r < 8; ++r) {
            const float mn = fminf(fminf(w0[r], w1[r]), w2[r]);  // v_min3_f32
            const bool upd = (mn >= 0.0f) && (zp[r] > bestZ[r]);
            bestZ[r] = upd ? zp[r] : bestZ[r];
            bestL[r] = upd ? lam   : bestL[r];
        }
    }

    // Cross-lane max over the 16 face columns (halves stay independent:
    // lanes 0-15 are pixel M=r, lanes 16-31 are pixel M=r+8).
    #pragma unroll
    for (int r = 0; r < 8; ++r) {
        float z = bestZ[r], l = bestL[r];
        #pragma unroll
        for (int msk = 8; msk >= 1; msk >>= 1) {
            float oz = __shfl_xor(z, msk, 32);
            float ol = __shfl_xor(l, msk, 32);
            if (oz > z) { z = oz; l = ol; }
        }
        if ((lane & 15) == 0) {
            const int M = r + ((lane >> 4) << 3);
            const int pp = pixBase + M;
            const float red = (z > BIG_NEG * 0.5f) ? l : 0.0f;
            out[3 * pp + 0] = red;
            out[3 * pp + 1] = 0.0f;
            out[3 * pp + 2] = 0.0f;
        }
    }
}

extern "C" void kernel_launch(void* const* d_in, const int* in_sizes, int n_in,
                              void* d_out, int out_size, void* d_ws, size_t ws_size,
                              hipStream_t stream) {
    const float* verts = (const float*)d_in[0];
    const int*   faces = (const int*)d_in[1];
    const float* Robj  = (const float*)d_in[2];
    const float* Tobj  = (const float*)d_in[3];
    const float* Kc    = (const float*)d_in[4];
    const float* Rw2c  = (const float*)d_in[5];
    const float* tw2c  = (const float*)d_in[6];
    float* wsf = (float*)d_ws;
    float* out = (float*)d_out;

    const int nv = in_sizes[0] / 3;
    const int nf = in_sizes[1] / 3;

    geom_kernel<<<1, 512, 0, stream>>>(verts, faces, Robj, Tobj, Kc, Rw2c, tw2c,
                                       wsf, nv, nf);
    const int npix = H_IMG * W_IMG;              // 76800
    const int blocks = npix / (8 * 16);          // 600 blocks, 8 waves x 16 px
    raster_kernel<<<blocks, 256, 0, stream>>>(wsf, out);
}